// SelfAttention_62869731279483
// MI455X (gfx1250) — compile-verified
//
#include <hip/hip_runtime.h>
#include <math.h>
#include <stdint.h>

// SAGAN self-attention for MI455X (gfx1250), fp32 WMMA 16x16x4 path.
// B=8, L=2048, F=512, C=64. Bandwidth-bound on the attn [B,L,L] write
// (134 MB of 200 MB total) -> stream it via GLOBAL_STORE_ASYNC_FROM_LDS_B128.

typedef float v2f __attribute__((ext_vector_type(2)));
typedef float v8f __attribute__((ext_vector_type(8)));

#define B_ 8
#define L_ 2048
#define F_ 512
#define C_ 64

__device__ __forceinline__ v8f wmma4(v2f a, v2f b, v8f c) {
  // 8 args: (neg_a, A, neg_b, B, c_mod, C, reuse_a, reuse_b)
  return __builtin_amdgcn_wmma_f32_16x16x4_f32(false, a, false, b, (short)0, c,
                                               false, false);
}

__device__ __forceinline__ v2f ld2(const float* p) { return *(const v2f*)p; }

// Async LDS -> global store, 16B per lane, tracked by ASYNCcnt.
__device__ __forceinline__ void async_store_b128(const float* gaddr,
                                                 uint32_t lds_off) {
  asm volatile("global_store_async_from_lds_b128 %0, %1, off" ::"v"(gaddr),
               "v"(lds_off)
               : "memory");
}

// ---------------------------------------------------------------------------
// K1: fused q/k/v projections.  [16384 x 512] @ [512 x 64] x3.
// One wave -> 16 rows x 64 cols x 3 outputs (12 accumulators), K=512.
// ---------------------------------------------------------------------------
__global__ __launch_bounds__(256) void qkv_kernel(
    const float* __restrict__ x, const float* __restrict__ Wq,
    const float* __restrict__ Wk, const float* __restrict__ Wv,
    float* __restrict__ q, float* __restrict__ k, float* __restrict__ v) {
  const int lane = threadIdx.x & 31;
  const int wave = threadIdx.x >> 5;
  const int lh = lane & 15;   // row / col within tile
  const int hi = lane >> 4;   // K-pair select
  const int rowTile = (blockIdx.x * 8 + wave) * 16;

  const v8f vz = {};
  v8f accq[4], acck[4], accv[4];
  for (int i = 0; i < 4; ++i) { accq[i] = vz; acck[i] = vz; accv[i] = vz; }

  const float* xrow = x + (size_t)(rowTile + lh) * F_;
  for (int kk = 0; kk < F_ / 4; ++kk) {
    const int kof = 4 * kk + 2 * hi;
    v2f a = ld2(xrow + kof);  // A(M=lh, K=kof..kof+1), contiguous 8B
#pragma unroll
    for (int nt = 0; nt < 4; ++nt) {
      const int col = nt * 16 + lh;
      v2f bq = {Wq[kof * C_ + col], Wq[(kof + 1) * C_ + col]};
      v2f bk = {Wk[kof * C_ + col], Wk[(kof + 1) * C_ + col]};
      v2f bv = {Wv[kof * C_ + col], Wv[(kof + 1) * C_ + col]};
      accq[nt] = wmma4(a, bq, accq[nt]);
      acck[nt] = wmma4(a, bk, acck[nt]);
      accv[nt] = wmma4(a, bv, accv[nt]);
    }
  }
#pragma unroll
  for (int nt = 0; nt < 4; ++nt)
#pragma unroll
    for (int r = 0; r < 8; ++r) {
      const size_t idx = (size_t)(rowTile + r + 8 * hi) * C_ + nt * 16 + lh;
      q[idx] = accq[nt][r];
      k[idx] = acck[nt][r];
      v[idx] = accv[nt][r];
    }
}

// ---------------------------------------------------------------------------
// K2: scores = q@k^T, softmax over QUERY axis l (per column m), write attn,
// and afm[c,m] = sum_l v[l,c]*attn[l,m].  One wave owns a 16-wide m strip and
// sweeps all l twice (recompute; q/k/v live in L2).  attn tiles are staged in
// double-buffered LDS slots and streamed out with async LDS->global b128
// stores that overlap the afm WMMAs.
// ---------------------------------------------------------------------------
__global__ __launch_bounds__(256) void attn_kernel(
    const float* __restrict__ q, const float* __restrict__ k,
    const float* __restrict__ v, const float* __restrict__ mask,
    float* __restrict__ attn, float* __restrict__ afm) {
  __shared__ float sAttn[8][2][256];  // per-wave double-buffered 16x16 tile
  const int lane = threadIdx.x & 31;
  const int wave = threadIdx.x >> 5;
  const int lh = lane & 15;
  const int hi = lane >> 4;
  const int b = blockIdx.x >> 4;   // 16 m-tiles per batch
  const int tile = blockIdx.x & 15;
  const int mBase = (tile * 8 + wave) * 16;

  const float* qb = q + (size_t)b * L_ * C_;
  const float* kbp = k + (size_t)b * L_ * C_;
  const float* vb = v + (size_t)b * L_ * C_;
  float* attnb = attn + (size_t)b * L_ * L_;
  float* afmb = afm + (size_t)b * C_ * L_;

  // B operand (k^T for this m strip) is loop-invariant: keep in registers.
  v2f kreg[16];
#pragma unroll
  for (int kk = 0; kk < 16; ++kk)
    kreg[kk] = ld2(kbp + (size_t)(mBase + lh) * C_ + 4 * kk + 2 * hi);

  const v8f vz = {};

  // Pass 1: per-column (m) running max and sum of exp over all l.
  float runMax = -INFINITY, runSum = 0.f;
  for (int lt = 0; lt < L_ / 16; ++lt) {
    const int lBase = lt * 16;
    const float* qrow = qb + (size_t)(lBase + lh) * C_;
    v8f s = vz;
#pragma unroll
    for (int kk = 0; kk < 16; ++kk)
      s = wmma4(ld2(qrow + 4 * kk + 2 * hi), kreg[kk], s);
    float tm = s[0];
#pragma unroll
    for (int r = 1; r < 8; ++r) tm = fmaxf(tm, s[r]);
    tm = fmaxf(tm, __shfl_xor(tm, 16, 32));  // fold lane halves: column max
    const float newMax = fmaxf(runMax, tm);
    const float scale = expf(runMax - newMax);  // 0 on first iter (-inf)
    float ts = 0.f;
#pragma unroll
    for (int r = 0; r < 8; ++r) ts += expf(s[r] - newMax);
    ts += __shfl_xor(ts, 16, 32);
    runSum = runSum * scale + ts;
    runMax = newMax;
  }
  const float invSum = 1.f / runSum;

  // Pass 2: recompute scores, stage attn tile in LDS, async-stream it to
  // global, and accumulate afm from the same LDS tile.
  v8f afmAcc[4];
  for (int i = 0; i < 4; ++i) afmAcc[i] = vz;
  const int rowSel = lane >> 1;   // 0..15: tile row handled by this lane
  const int halfSel = lane & 1;   // which 8-float half of the row
  for (int lt = 0; lt < L_ / 16; ++lt) {
    const int lBase = lt * 16;
    const int par = lt & 1;
    const float* qrow = qb + (size_t)(lBase + lh) * C_;
    v8f s = vz;
#pragma unroll
    for (int kk = 0; kk < 16; ++kk)
      s = wmma4(ld2(qrow + 4 * kk + 2 * hi), kreg[kk], s);
    // Allow only the previous iteration's 2 async stores to be outstanding:
    // drains the stores that read the tile slot we are about to overwrite.
    asm volatile("s_wait_asynccnt 0x2" ::: "memory");
#pragma unroll
    for (int r = 0; r < 8; ++r) {
      const float a = expf(s[r] - runMax) * invSum;
      sAttn[wave][par][(r + 8 * hi) * 16 + lh] = a;  // D-layout -> LDS tile
    }
    __syncthreads();  // uniform; orders intra-wave cross-lane LDS store->load
    // Tile resident in LDS: stream it to global without a VGPR round-trip.
    asm volatile("s_wait_dscnt 0x0" ::: "memory");
    {
      const float* gp =
          attnb + (size_t)(lBase + rowSel) * L_ + mBase + halfSel * 8;
      const uint32_t lp = (uint32_t)(uintptr_t)&sAttn[wave][par]
                                                     [rowSel * 16 + halfSel * 8];
      async_store_b128(gp, lp);
      async_store_b128(gp + 4, lp + 16);
    }
    // afm: A = v^T tile (M=c, K=l), B = attn tile from LDS (K=l, N=m).
#pragma unroll
    for (int kk2 = 0; kk2 < 4; ++kk2) {
      const int lof = 4 * kk2 + 2 * hi;
      v2f battn = {sAttn[wave][par][lof * 16 + lh],
                   sAttn[wave][par][(lof + 1) * 16 + lh]};
#pragma unroll
      for (int ct = 0; ct < 4; ++ct) {
        v2f av = {vb[(size_t)(lBase + lof) * C_ + ct * 16 + lh],
                  vb[(size_t)(lBase + lof + 1) * C_ + ct * 16 + lh]};
        afmAcc[ct] = wmma4(av, battn, afmAcc[ct]);
      }
    }
  }
  const float mval = mask[(size_t)b * L_ + mBase + lh];  // afm *= mask
#pragma unroll
  for (int ct = 0; ct < 4; ++ct)
#pragma unroll
    for (int r = 0; r < 8; ++r)
      afmb[(size_t)(ct * 16 + r + 8 * hi) * L_ + mBase + lh] =
          afmAcc[ct][r] * mval;
  // S_ENDPGM performs an implicit wait-idle; outstanding async stores drain.
}

// ---------------------------------------------------------------------------
// K3: out = x + gamma * (afm^T @ Wo).  [16384 x 64] @ [64 x 512] + residual.
// ---------------------------------------------------------------------------
__global__ __launch_bounds__(256) void out_kernel(
    const float* __restrict__ x, const float* __restrict__ afm,
    const float* __restrict__ Wo, const float* __restrict__ gamma,
    float* __restrict__ out) {
  const int lane = threadIdx.x & 31;
  const int wave = threadIdx.x >> 5;
  const int lh = lane & 15;
  const int hi = lane >> 4;
  const int task = blockIdx.x * 8 + wave;
  const int colGroup = task & 7;         // 8 groups of 64 output cols
  const int rowTile = (task >> 3) * 16;  // global position row (b*L+l)
  const int b = rowTile / L_;
  const int m = rowTile % L_;
  const float* afmb = afm + (size_t)b * C_ * L_;
  const float g = gamma[0];

  const v8f vz = {};
  v8f acc[4];
  for (int i = 0; i < 4; ++i) acc[i] = vz;
#pragma unroll
  for (int kk = 0; kk < 16; ++kk) {
    const int kof = 4 * kk + 2 * hi;
    // A(M=m+lh, K=c) = afm[c, m+lh] (transposed read, stride L)
    v2f a = {afmb[(size_t)kof * L_ + m + lh],
             afmb[(size_t)(kof + 1) * L_ + m + lh]};
#pragma unroll
    for (int nt = 0; nt < 4; ++nt) {
      const int col = colGroup * 64 + nt * 16 + lh;
      v2f bw = {Wo[kof * F_ + col], Wo[(kof + 1) * F_ + col]};
      acc[nt] = wmma4(a, bw, acc[nt]);
    }
  }
#pragma unroll
  for (int nt = 0; nt < 4; ++nt)
#pragma unroll
    for (int r = 0; r < 8; ++r) {
      const size_t idx =
          (size_t)(rowTile + r + 8 * hi) * F_ + colGroup * 64 + nt * 16 + lh;
      out[idx] = x[idx] + g * acc[nt][r];
    }
}

// ---------------------------------------------------------------------------
extern "C" void kernel_launch(void* const* d_in, const int* in_sizes, int n_in,
                              void* d_out, int out_size, void* d_ws,
                              size_t ws_size, hipStream_t stream) {
  (void)in_sizes; (void)n_in; (void)out_size; (void)ws_size;
  const float* x = (const float*)d_in[0];
  const float* Wq = (const float*)d_in[1];
  const float* Wk = (const float*)d_in[2];
  const float* Wv = (const float*)d_in[3];
  const float* Wo = (const float*)d_in[4];
  const float* gamma = (const float*)d_in[5];
  const float* mask = (const float*)d_in[6];

  float* out = (float*)d_out;                    // B*L*F
  float* attn = out + (size_t)B_ * L_ * F_;      // B*L*L (second output)

  float* q = (float*)d_ws;                       // 4 MB each
  float* k = q + (size_t)B_ * L_ * C_;
  float* v = k + (size_t)B_ * L_ * C_;
  float* afm = v + (size_t)B_ * L_ * C_;         // [B, C, L]

  qkv_kernel<<<dim3((B_ * L_) / (16 * 8)), dim3(256), 0, stream>>>(
      x, Wq, Wk, Wv, q, k, v);
  attn_kernel<<<dim3(B_ * (L_ / 128)), dim3(256), 0, stream>>>(
      q, k, v, mask, attn, afm);
  out_kernel<<<dim3(((B_ * L_ / 16) * (F_ / 64)) / 8), dim3(256), 0, stream>>>(
      x, afm, Wo, gamma, out);
}